// GDMLTorchPredict_54580444398239
// MI455X (gfx1250) — compile-verified
//
#include <hip/hip_runtime.h>
#include <math.h>

typedef __attribute__((ext_vector_type(2))) float v2f;
typedef __attribute__((ext_vector_type(8))) float v8f;

// Problem constants (from reference)
constexpr int   kM  = 64;     // molecules
constexpr int   kAt = 24;     // atoms
constexpr int   kP  = 276;    // descriptors = 24*23/2
constexpr int   kPP = 288;    // padded P (multiple of 16)
constexpr int   kT  = 6000;   // training rows
constexpr int   kNS = 10;     // K-splits for phase-2 GEMM
constexpr float kQ   = 0.22360679774997896f;  // sqrt(5)/SIG, SIG=10
constexpr float kC0  = 1.0f / 60.0f;          // 5/(3*SIG^2)
constexpr float kSTD = 1.0f;
constexpr float kC   = 0.0f;

__device__ __forceinline__ v8f wmma_f32(v2f a, v2f b, v8f c) {
  // D = A(16x4,f32) x B(4x16,f32) + C(16x16,f32)
  return __builtin_amdgcn_wmma_f32_16x16x4_f32(false, a, false, b, (short)0, c,
                                               false, false);
}

// ---------------------------------------------------------------------------
// Kernel 1: descriptors xs[m][p] = 1/|r_i - r_j| (tril pairs), padded to 288,
// plus nm[m] = ||xs_m||^2. One block per molecule.
// ---------------------------------------------------------------------------
__global__ void k_prep(const float* __restrict__ Rs, float* __restrict__ xs,
                       float* __restrict__ nm) {
  __shared__ float red[kPP];
  const int m = blockIdx.x;
  const int p = threadIdx.x;
  float x = 0.0f;
  if (p < kP) {
    // invert triangular index: p = i*(i-1)/2 + j, 0 <= j < i
    int i = (int)((1.0f + sqrtf(1.0f + 8.0f * (float)p)) * 0.5f);
    while (i * (i - 1) / 2 > p) --i;
    while ((i + 1) * i / 2 <= p) ++i;
    const int j = p - i * (i - 1) / 2;
    const float* ri = Rs + (m * kAt + i) * 3;
    const float* rj = Rs + (m * kAt + j) * 3;
    const float dx = ri[0] - rj[0];
    const float dy = ri[1] - rj[1];
    const float dz = ri[2] - rj[2];
    x = 1.0f / sqrtf(dx * dx + dy * dy + dz * dz);
  }
  xs[m * kPP + p] = x;
  red[p] = x * x;
  __syncthreads();
  if (p == 0) {
    float s = 0.0f;
    for (int q = 0; q < kP; ++q) s += red[q];  // fixed order: deterministic
    nm[m] = s;
  }
}

// ---------------------------------------------------------------------------
// Kernel 2: per training row t: nt[t]=||xs_train_t||^2, ct[t]=xs_train_t.Jx_t
// One wave per row.
// ---------------------------------------------------------------------------
__global__ void k_tstats(const float* __restrict__ xt, const float* __restrict__ jx,
                         float* __restrict__ nt, float* __restrict__ ct) {
  const int wave = (blockIdx.x * blockDim.x + threadIdx.x) >> 5;
  const int lane = threadIdx.x & 31;
  if (wave >= kT) return;
  const float* xr = xt + wave * kP;
  const float* jr = jx + wave * kP;
  float s1 = 0.0f, s2 = 0.0f;
  for (int p = lane; p < kP; p += 32) {
    const float v = xr[p];
    s1 += v * v;
    s2 += v * jr[p];
  }
  for (int off = 16; off > 0; off >>= 1) {
    s1 += __shfl_down(s1, off);
    s2 += __shfl_down(s2, off);
  }
  if (lane == 0) { nt[wave] = s1; ct[wave] = s2; }
}

// ---------------------------------------------------------------------------
// Kernel 3 (WMMA): phase-1 GEMMs G1=xs.xs_train^T, G2=xs.Jx^T fused with the
// kernel evaluation. One wave per 16x16 (m,t) tile; K=276 in steps of 4.
// Writes W[m,t]=exp_xs*dot, E1[m,t]=exp1, D[m,t]=dot.
// ---------------------------------------------------------------------------
__global__ void k_phase1(const float* __restrict__ xs, const float* __restrict__ xt,
                         const float* __restrict__ jx, const float* __restrict__ nm,
                         const float* __restrict__ nt, const float* __restrict__ ct,
                         float* __restrict__ Wm, float* __restrict__ E1,
                         float* __restrict__ Dm) {
  const int tile = blockIdx.x * (blockDim.x >> 5) + (threadIdx.x >> 5);
  const int ntile = 4 * (kT / 16);  // 1500
  if (tile >= ntile) return;        // wave-uniform: EXEC all-1 inside
  const int m0 = (tile & 3) * 16;
  const int t0 = (tile >> 2) * 16;
  const int lane = threadIdx.x & 31;
  const int nl = lane & 15;
  const int kh = (lane >> 4) << 1;  // K sub-offset 0 or 2

  const float* arow  = xs + (m0 + nl) * kPP + kh;  // A: lane = row m
  const float* b1row = xt + (t0 + nl) * kP + kh;   // B: lane = col t
  const float* b2row = jx + (t0 + nl) * kP + kh;

  v8f g1 = {}; v8f g2 = {};
  for (int k0 = 0; k0 < kP; k0 += 4) {
    const v2f a  = *(const v2f*)(arow + k0);
    const v2f b1 = *(const v2f*)(b1row + k0);
    const v2f b2 = *(const v2f*)(b2row + k0);
    g1 = wmma_f32(a, b1, g1);
    g2 = wmma_f32(a, b2, g2);
  }

  const int t = t0 + nl;
  const float ntv = nt[t];
  const float ctv = ct[t];
  const int mbase = m0 + ((lane >> 4) << 3);
  for (int r = 0; r < 8; ++r) {
    const int m = mbase + r;
    const float d2 = kQ * kQ * fmaxf(nm[m] - 2.0f * g1[r] + ntv, 0.0f);
    const float dist = sqrtf(d2);
    const float dotv = kQ * (g2[r] - ctv);
    const float ex = kC0 * __expf(-dist);
    const float e1 = ex * (1.0f + dist);
    Wm[m * kT + t] = ex * dotv;
    E1[m * kT + t] = e1;
    Dm[m * kT + t] = dotv;
  }
}

// ---------------------------------------------------------------------------
// Kernel 4: per-molecule row reductions: Es[m] = sum_t E1*D / q + C (to d_out)
// and S[m] = sum_t W (needed for F1). Deterministic strided + tree reduce.
// ---------------------------------------------------------------------------
__global__ void k_rowred(const float* __restrict__ Wm, const float* __restrict__ E1,
                         const float* __restrict__ Dm, float* __restrict__ Es_out,
                         float* __restrict__ S) {
  __shared__ float se[256];
  __shared__ float sw[256];
  const int m = blockIdx.x, tid = threadIdx.x;
  float a = 0.0f, b = 0.0f;
  for (int t = tid; t < kT; t += 256) {
    a += E1[m * kT + t] * Dm[m * kT + t];
    b += Wm[m * kT + t];
  }
  se[tid] = a; sw[tid] = b;
  __syncthreads();
  for (int s = 128; s > 0; s >>= 1) {
    if (tid < s) { se[tid] += se[tid + s]; sw[tid] += sw[tid + s]; }
    __syncthreads();
  }
  if (tid == 0) {
    Es_out[m] = se[0] * (1.0f / kQ) * kSTD + kC;
    S[m] = sw[0];
  }
}

// ---------------------------------------------------------------------------
// Kernel 5 (WMMA): phase-2 GEMMs F1b = W @ xs_train, F2 = E1 @ Jx over K=6000,
// split into kNS K-slices for parallelism; partials to workspace slabs.
// One wave per (split, 16m x 16p) tile. 720 tiles exactly.
// ---------------------------------------------------------------------------
__global__ void k_phase2(const float* __restrict__ Wm, const float* __restrict__ E1,
                         const float* __restrict__ xt, const float* __restrict__ jx,
                         float* __restrict__ F1p, float* __restrict__ F2p) {
  const int tile = blockIdx.x * (blockDim.x >> 5) + (threadIdx.x >> 5);
  const int split = tile / 72;
  const int rem = tile % 72;
  const int m0 = (rem & 3) * 16;
  const int p0 = (rem >> 2) * 16;
  const int lane = threadIdx.x & 31;
  const int nl = lane & 15;
  const int kh = (lane >> 4) << 1;
  const int pidx = (p0 + nl < kP) ? (p0 + nl) : (kP - 1);  // clamp pad cols
  const int kbeg = split * (kT / kNS);
  const int kend = kbeg + (kT / kNS);

  const float* aWrow = Wm + (m0 + nl) * kT + kh;
  const float* aErow = E1 + (m0 + nl) * kT + kh;

  v8f f1 = {}; v8f f2 = {};
  for (int k0 = kbeg; k0 < kend; k0 += 4) {
    const v2f aW = *(const v2f*)(aWrow + k0);
    const v2f aE = *(const v2f*)(aErow + k0);
    const int krow = k0 + kh;
    v2f b1, b2;
    b1.x = xt[krow * kP + pidx];       b1.y = xt[(krow + 1) * kP + pidx];
    b2.x = jx[krow * kP + pidx];       b2.y = jx[(krow + 1) * kP + pidx];
    f1 = wmma_f32(aW, b1, f1);
    f2 = wmma_f32(aE, b2, f2);
  }

  const int p = p0 + nl;  // p < 288 always (pad region holds garbage, unused)
  const int mbase = m0 + ((lane >> 4) << 3);
  for (int r = 0; r < 8; ++r) {
    const int m = mbase + r;
    F1p[(split * kM + m) * kPP + p] = f1[r];
    F2p[(split * kM + m) * kPP + p] = f2[r];
  }
}

// ---------------------------------------------------------------------------
// Kernel 6: Fs_x[m,p] = (q*xs*S[m] - q*F1b - F2) * xs^3, then scatter pair
// forces onto atoms: Fs[m,b] = sum_a (R_a - R_b) * mult(a,b). One block per m.
// ---------------------------------------------------------------------------
__global__ void k_final(const float* __restrict__ Rs, const float* __restrict__ xs,
                        const float* __restrict__ S, const float* __restrict__ F1p,
                        const float* __restrict__ F2p, float* __restrict__ Fs_out) {
  __shared__ float fsx[kPP];
  const int m = blockIdx.x, tid = threadIdx.x;
  if (tid < kP) {
    float a = 0.0f, b = 0.0f;
    for (int s = 0; s < kNS; ++s) {  // fixed order: deterministic
      a += F1p[(s * kM + m) * kPP + tid];
      b += F2p[(s * kM + m) * kPP + tid];
    }
    const float x = xs[m * kPP + tid];
    const float F1 = kQ * x * S[m] - kQ * a;
    fsx[tid] = (F1 - b) * x * x * x;
  } else if (tid < kPP) {
    fsx[tid] = 0.0f;
  }
  __syncthreads();
  if (tid < kAt * 3) {
    const int bA = tid / 3, c = tid % 3;
    const float* Rm = Rs + m * kAt * 3;
    const float rb = Rm[bA * 3 + c];
    float acc = 0.0f;
    for (int a = 0; a < kAt; ++a) {
      if (a == bA) continue;
      const int i = (a > bA) ? a : bA;
      const int j = (a > bA) ? bA : a;
      const int pp = i * (i - 1) / 2 + j;
      acc += (Rm[a * 3 + c] - rb) * fsx[pp];
    }
    Fs_out[m * kAt * 3 + tid] = acc * kSTD;
  }
}

// ---------------------------------------------------------------------------
extern "C" void kernel_launch(void* const* d_in, const int* in_sizes, int n_in,
                              void* d_out, int out_size, void* d_ws, size_t ws_size,
                              hipStream_t stream) {
  (void)in_sizes; (void)n_in; (void)out_size; (void)ws_size;
  const float* Rs = (const float*)d_in[0];  // [64,24,3]
  const float* xt = (const float*)d_in[1];  // [6000,276]
  const float* jx = (const float*)d_in[2];  // [6000,276]
  float* out = (float*)d_out;               // [64] Es ++ [64*24*3] Fs

  float* ws  = (float*)d_ws;
  float* xs  = ws;                    // 64*288
  float* nm  = xs + kM * kPP;         // 64
  float* nt  = nm + kM;               // 6000
  float* ct  = nt + kT;               // 6000
  float* S   = ct + kT;               // 64
  float* Wm  = S + kM;                // 64*6000
  float* E1  = Wm + kM * kT;          // 64*6000
  float* Dm  = E1 + kM * kT;          // 64*6000
  float* F1p = Dm + kM * kT;          // kNS*64*288
  float* F2p = F1p + kNS * kM * kPP;  // kNS*64*288

  k_prep<<<kM, kPP, 0, stream>>>(Rs, xs, nm);
  k_tstats<<<(kT * 32 + 255) / 256, 256, 0, stream>>>(xt, jx, nt, ct);

  const int tiles1 = 4 * (kT / 16);  // 1500 waves, 8 waves/block
  k_phase1<<<(tiles1 + 7) / 8, 256, 0, stream>>>(xs, xt, jx, nm, nt, ct, Wm, E1, Dm);

  k_rowred<<<kM, 256, 0, stream>>>(Wm, E1, Dm, out, S);

  const int tiles2 = kNS * 4 * 18;   // 720 waves = 90 blocks * 8 waves
  k_phase2<<<tiles2 / 8, 256, 0, stream>>>(Wm, E1, xt, jx, F1p, F2p);

  k_final<<<kM, kPP, 0, stream>>>(Rs, xs, S, F1p, F2p, out + kM);
}